// APoTQuantizer_3865470566880
// MI455X (gfx1250) — compile-verified
//
#include <hip/hip_runtime.h>

typedef float v4f __attribute__((ext_vector_type(4)));

#define BLOCK 256  // 8 wave32 waves per workgroup

#if defined(__HIP_DEVICE_COMPILE__) && __has_builtin(__builtin_amdgcn_global_load_async_to_lds_b32)
#define HAVE_ASYNC_COPY 1
typedef __attribute__((address_space(1))) int as1_int;  // '__device__ int *'
typedef __attribute__((address_space(3))) int as3_int;  // '__shared__ int *'
#endif

__device__ __forceinline__ void wait_asynccnt0() {
#if defined(__HIP_DEVICE_COMPILE__)
#if __has_builtin(__builtin_amdgcn_s_wait_asynccnt)
    __builtin_amdgcn_s_wait_asynccnt(0);
#else
    asm volatile("s_wait_asynccnt 0x0" ::: "memory");
#endif
#endif
}

// Nearest-level snap == searchsorted(midpoints, x, 'left') with strict '<'
// (preserves the reference's tie-to-left). Table layout: s_tab[k] = mid[k-1]
// for k=1..P (+INF padded), s_tab[0] = sentinel (never probed: pos|bit >= 1).
// Search runs on BYTE offsets: pos/bit are disjoint pow2 masks, so OR works
// on byte addresses too. Probe 1 (root of the tree, s_tab[P/2]) is
// loop-invariant and hoisted into a register.
template <int P>
__global__ __launch_bounds__(BLOCK) void apot_quant_kernel(
    const float* __restrict__ x,
    const float* __restrict__ alpha,
    const float* __restrict__ levels,
    float* __restrict__ out,
    int n, int n4, int L)
{
    __shared__ float s_levels[P];     // levels, +INF padded
    __shared__ float s_tab[P + 1];    // shifted midpoints, +INF padded

    const float INF = __builtin_inff();

#ifdef HAVE_ASYNC_COPY
    // CDNA5 async bulk copy: global levels[] -> LDS, tracked by ASYNCcnt.
    for (int i = threadIdx.x; i < L; i += BLOCK) {
        __builtin_amdgcn_global_load_async_to_lds_b32(
            (as1_int*)(levels + i),
            (as3_int*)&s_levels[i],
            0, 0);
    }
    for (int i = L + (int)threadIdx.x; i < P; i += BLOCK)
        s_levels[i] = INF;                     // pad region
    if (threadIdx.x == 0) s_tab[0] = -INF;     // sentinel (unread)
    wait_asynccnt0();
    __syncthreads();
    // Derive shifted midpoint table from the LDS level copy.
    for (int i = threadIdx.x; i < P; i += BLOCK) {
        float lv = s_levels[i];
        float nx = (i + 1 < P) ? s_levels[i + 1] : INF;
        s_tab[i + 1] = (i + 1 < L) ? 0.5f * (lv + nx) : INF;
    }
    __syncthreads();
#else
    // Fallback: build both tables straight from global memory.
    for (int i = threadIdx.x; i < P; i += BLOCK) {
        float lv = (i < L) ? levels[i] : INF;
        float nx = (i + 1 < L) ? levels[i + 1] : INF;
        s_levels[i]  = lv;
        s_tab[i + 1] = (i + 1 < L) ? 0.5f * (lv + nx) : INF;
    }
    if (threadIdx.x == 0) s_tab[0] = -INF;
    __syncthreads();
#endif

    const float a     = fabsf(alpha[0]) + 1e-8f;  // uniform -> scalar load
    const float inv_a = 1.0f / a;

    const char* tab = (const char*)s_tab;      // LDS bases fold into ds offsets
    const char* lev = (const char*)s_levels;

    // Loop-invariant root probe (pos=0 -> always s_tab[P/2]).
    const unsigned rootB = (unsigned)(P >> 1) * 4u;
    const float m_root = s_tab[P >> 1];

    const v4f* __restrict__ x4 = reinterpret_cast<const v4f*>(x);
    v4f* __restrict__ o4       = reinterpret_cast<v4f*>(out);

    const int stride = gridDim.x * BLOCK;

    for (int idx = blockIdx.x * BLOCK + threadIdx.x; idx < n4; idx += stride) {
        if (idx + stride < n4)
            __builtin_prefetch(reinterpret_cast<const char*>(x4 + idx + stride), 0, 1);

        v4f v = __builtin_nontemporal_load(x4 + idx);  // 128-bit streaming load

        float xn[4];
        unsigned posB[4];
#pragma unroll
        for (int c = 0; c < 4; ++c) {
            xn[c]   = fminf(1.0f, fmaxf(-1.0f, v[c] * inv_a));
            posB[c] = (m_root < xn[c]) ? rootB : 0u;   // hoisted probe 1
        }

        // Remaining log2(P)-1 probes: byte-offset binary search, 4-way DS ILP
#pragma unroll
        for (unsigned bitB = (unsigned)(P >> 2) * 4u; bitB >= 4u; bitB >>= 1) {
#pragma unroll
            for (int c = 0; c < 4; ++c) {
                unsigned nxtB = posB[c] | bitB;
                float m = *(const float*)(tab + nxtB);
                posB[c] = (m < xn[c]) ? nxtB : posB[c];
            }
        }

        v4f r;
#pragma unroll
        for (int c = 0; c < 4; ++c)
            r[c] = a * (*(const float*)(lev + posB[c]));

        __builtin_nontemporal_store(r, o4 + idx);  // 128-bit NT store
    }

    // Scalar tail for n % 4 != 0 (no-op for the reference shape).
    const int tail_start = n4 << 2;
    const int t = blockIdx.x * BLOCK + threadIdx.x;
    if (tail_start + t < n) {
        float xv = fminf(1.0f, fmaxf(-1.0f, x[tail_start + t] * inv_a));
        unsigned posB = (m_root < xv) ? rootB : 0u;
#pragma unroll
        for (unsigned bitB = (unsigned)(P >> 2) * 4u; bitB >= 4u; bitB >>= 1) {
            unsigned nxtB = posB | bitB;
            float m = *(const float*)(tab + nxtB);
            posB = (m < xv) ? nxtB : posB;
        }
        out[tail_start + t] = a * (*(const float*)(lev + posB));
    }
}

extern "C" void kernel_launch(void* const* d_in, const int* in_sizes, int n_in,
                              void* d_out, int out_size, void* d_ws, size_t ws_size,
                              hipStream_t stream) {
    const float* x      = (const float*)d_in[0];
    const float* alpha  = (const float*)d_in[1];
    const float* levels = (const float*)d_in[2];
    float* out          = (float*)d_out;

    const int n  = in_sizes[0];
    const int L  = in_sizes[2];   // 511 for the reference APoT(8-bit, n=2, signed)
    const int n4 = n >> 2;

    int need = (n4 + BLOCK - 1) / BLOCK;
    if (need < 1) need = 1;
    int blocks = need < 4096 ? need : 4096;

    // P = smallest power of two >= L (guarantees pos stays in-bounds when
    // x_norm == 1 exceeds every midpoint).
    int P = 256;
    while (P < L && P < 4096) P <<= 1;

    switch (P) {
    case 256:
        apot_quant_kernel<256><<<blocks, BLOCK, 0, stream>>>(x, alpha, levels, out, n, n4, L);
        break;
    case 512:
        apot_quant_kernel<512><<<blocks, BLOCK, 0, stream>>>(x, alpha, levels, out, n, n4, L);
        break;
    case 1024:
        apot_quant_kernel<1024><<<blocks, BLOCK, 0, stream>>>(x, alpha, levels, out, n, n4, L);
        break;
    case 2048:
        apot_quant_kernel<2048><<<blocks, BLOCK, 0, stream>>>(x, alpha, levels, out, n, n4, L);
        break;
    default:
        apot_quant_kernel<4096><<<blocks, BLOCK, 0, stream>>>(x, alpha, levels, out, n, n4, L);
        break;
    }
}